// RWLayer_88072599371933
// MI455X (gfx1250) — compile-verified
//
#include <hip/hip_runtime.h>

// GAT-style attention (N=6144, D=128, H=8, HID=16) for MI455X / gfx1250.
//
// Roofline: mandatory HBM traffic = adj (151MB, L2-resident after first read)
// + out (151MB) ~= 13-26us @ 23.3TB/s. Naive softmax costs >=2 v_exp per
// (h,i,j) = 600M+ transcendentals >> memory time. We factor the exps through
// the piecewise-linear LeakyReLU so the N*N inner loops are pure
// add/cmp/cndmask/fma (no transcendentals); exps happen only O(H*N) times.
// The small x@linear projection runs on fp32 WMMA (V_WMMA_F32_16X16X4_F32).

#define N_NODES 6144
#define DIM     128
#define HEADS   8
#define HID     16
#define SLOPE   0.2f

#define CH             128                     // rows staged in LDS per chunk
#define ISLICES        16                      // row slices (grid.y)
#define ROWS_PER_SLICE (N_NODES / ISLICES)     // 384

typedef float v2f __attribute__((ext_vector_type(2)));
typedef float v8f __attribute__((ext_vector_type(8)));

// ---------------------------------------------------------------------------
// Phase A: h = x @ linear[h]  (per head, M=6144 K=128 N=16) via fp32 WMMA,
// immediately contracted with att_src/att_tar -> s_src[h,n], s_tar[h,n].
// 8 waves per block = 8 heads; each wave owns a 16-row tile of x.
// A-tile layout (16x4 f32): lane m=lane&15, k = 2*(lane>>4) + vgpr.
// B-tile layout (4x16 f32): lane n=lane&15, k = 2*(lane>>4) + vgpr (mirror).
// C layout (16x16 f32): vgpr v, lane L -> M = v + 8*(L>>4), N = L&15.
// ---------------------------------------------------------------------------
__global__ __launch_bounds__(256) void proj_wmma_kernel(
    const float* __restrict__ x, const float* __restrict__ linear,
    const float* __restrict__ att_src, const float* __restrict__ att_tar,
    float* __restrict__ s_src, float* __restrict__ s_tar)
{
    const int tid  = threadIdx.x;
    const int h    = tid >> 5;           // wave id == head
    const int lane = tid & 31;
    const int r    = lane & 15;
    const int half = lane >> 4;
    const int m0   = blockIdx.x * 16;

    const float* xrow = x + (m0 + r) * DIM;
    const float* lb   = linear + h * DIM * HID;

    v8f c = {};
    for (int k = 0; k < DIM; k += 4) {
        const int ka = k + 2 * half;
        v2f a, b;
        a.x = xrow[ka];
        a.y = xrow[ka + 1];
        b.x = lb[(ka)     * HID + r];
        b.y = lb[(ka + 1) * HID + r];
        // D = A(16x4) * B(4x16) + C, full fp32
        c = __builtin_amdgcn_wmma_f32_16x16x4_f32(
                false, a, false, b, (short)0, c, false, false);
    }

    const float as = att_src[h * HID + r];
    const float at = att_tar[h * HID + r];

    #pragma unroll
    for (int v = 0; v < 8; ++v) {
        float ps = c[v] * as;   // dot over hidden dim e (= lane index r)
        float pt = c[v] * at;
        #pragma unroll
        for (int m = 1; m < 16; m <<= 1) {   // reduce within each 16-lane half
            ps += __shfl_xor(ps, m, 32);
            pt += __shfl_xor(pt, m, 32);
        }
        if (r == 0) {
            const int row = m0 + v + 8 * half;
            s_src[h * N_NODES + row] = ps;
            s_tar[h * N_NODES + row] = pt;
        }
    }
}

// ---------------------------------------------------------------------------
// Global max of s_tar per head (numerical reference point M_h).
// ---------------------------------------------------------------------------
__global__ __launch_bounds__(256) void maxred_kernel(
    const float* __restrict__ s_tar, float* __restrict__ M)
{
    __shared__ float red[256];
    const int h = blockIdx.x;
    float m = -1e30f;
    for (int i = threadIdx.x; i < N_NODES; i += 256)
        m = fmaxf(m, s_tar[h * N_NODES + i]);
    red[threadIdx.x] = m;
    __syncthreads();
    for (int s = 128; s > 0; s >>= 1) {
        if (threadIdx.x < s)
            red[threadIdx.x] = fmaxf(red[threadIdx.x], red[threadIdx.x + s]);
        __syncthreads();
    }
    if (threadIdx.x == 0) M[h] = red[0];
}

// ---------------------------------------------------------------------------
// Per-row factors  TP[i][h] = (t, P1=exp(t-Mh), P2=exp(0.2(t-Mh)), pad)  and
// per-col factors  CC[j] = { s[8], C1[8], C2[8] }  with m_j = LeakyReLU(Mh+s).
// All factors <= 1 -> no overflow anywhere; underflow only on negligible terms.
// ---------------------------------------------------------------------------
__global__ __launch_bounds__(256) void pexp_kernel(
    const float* __restrict__ s_src, const float* __restrict__ s_tar,
    const float* __restrict__ M,
    float* __restrict__ TP,    // [N][32] : (t,P1,P2,pad) x 8 heads (float4/head)
    float* __restrict__ CC)    // [N][32] : s[0..7], C1[8..15], C2[16..23]
{
    const int i = blockIdx.x * 256 + threadIdx.x;
    if (i >= N_NODES) return;
    #pragma unroll
    for (int h = 0; h < HEADS; ++h) {
        const float Mh = M[h];
        const float t  = s_tar[h * N_NODES + i];
        const float s  = s_src[h * N_NODES + i];
        const float dt = t - Mh;
        TP[i * 32 + h * 4 + 0] = t;
        TP[i * 32 + h * 4 + 1] = __expf(dt);
        TP[i * 32 + h * 4 + 2] = __expf(SLOPE * dt);
        TP[i * 32 + h * 4 + 3] = 0.0f;
        const float u = Mh + s;
        const float m = (u >= 0.0f) ? u : SLOPE * u;  // LeakyReLU: column max bound
        CC[i * 32 + h]      = s;
        CC[i * 32 + 8 + h]  = __expf(u - m);          // <= 1
        CC[i * 32 + 16 + h] = __expf(SLOPE * u - m);  // <= 1
    }
}

// ---------------------------------------------------------------------------
// Phase B: per-column partial sums over a row slice.
//   S1 = sum_{i: adj>0, t_i+s_j>=0} P1[i],  S2 = sum_{i: adj>0, neg} P2[i]
// Thread = one column j, 8 heads in registers; rows staged in LDS as float4.
// adj read is fully coalesced (128B/wave/row).
// ---------------------------------------------------------------------------
__global__ __launch_bounds__(256) void colsum_kernel(
    const int* __restrict__ adj, const float* __restrict__ TP,
    const float* __restrict__ CC,
    float* __restrict__ S1p, float* __restrict__ S2p)   // [ISLICES][H][N]
{
    __shared__ float4 lds[CH * 8];   // 16 KB
    const int j       = blockIdx.x * 256 + threadIdx.x;
    const int slice   = blockIdx.y;
    const int i0slice = slice * ROWS_PER_SLICE;

    float sj[8], a1[8], a2[8];
    #pragma unroll
    for (int h = 0; h < 8; ++h) {
        sj[h] = CC[j * 32 + h];
        a1[h] = 0.0f; a2[h] = 0.0f;
    }

    for (int c0 = 0; c0 < ROWS_PER_SLICE; c0 += CH) {
        const int i0 = i0slice + c0;
        __syncthreads();
        const float4* src = (const float4*)(TP + i0 * 32);
        for (int t = threadIdx.x; t < CH * 8; t += 256) lds[t] = src[t];
        __syncthreads();
        for (int ii = 0; ii < CH; ++ii) {
            const int   a    = adj[(i0 + ii) * N_NODES + j];
            const float adjf = (a > 0) ? 1.0f : 0.0f;
            #pragma unroll
            for (int h = 0; h < 8; ++h) {
                const float4 q = lds[ii * 8 + h];       // (t, P1, P2, pad)
                const float  v = q.x + sj[h];
                const bool pos = (v >= 0.0f);
                a1[h] = fmaf(adjf, pos ? q.y : 0.0f, a1[h]);
                a2[h] = fmaf(adjf, pos ? 0.0f : q.z, a2[h]);
            }
        }
    }
    #pragma unroll
    for (int h = 0; h < 8; ++h) {
        S1p[(slice * HEADS + h) * N_NODES + j] = a1[h];
        S2p[(slice * HEADS + h) * N_NODES + j] = a2[h];
    }
}

// Deterministic slice reduction (avoids float atomics).
__global__ __launch_bounds__(256) void slice_reduce_kernel(
    const float* __restrict__ S1p, const float* __restrict__ S2p,
    float* __restrict__ S1, float* __restrict__ S2)
{
    const int j = blockIdx.x * 256 + threadIdx.x;
    const int h = blockIdx.y;
    float a = 0.0f, b = 0.0f;
    for (int s = 0; s < ISLICES; ++s) {
        a += S1p[(s * HEADS + h) * N_NODES + j];
        b += S2p[(s * HEADS + h) * N_NODES + j];
    }
    S1[h * N_NODES + j] = a;
    S2[h * N_NODES + j] = b;
}

// ---------------------------------------------------------------------------
// Phase C: out[i,j] = adj ? (1/8) * sum_h (pos ? P1*R1 : P2*R2) : 0
// with R1 = C1/denom, R2 = C2/denom, denom = S1*C1 + S2*C2 (recomputed per
// thread from per-column data; cheap, keeps kernel count down).
// adj is L2-resident from phase B; stores are coalesced.
// ---------------------------------------------------------------------------
__global__ __launch_bounds__(256) void out_kernel(
    const int* __restrict__ adj, const float* __restrict__ TP,
    const float* __restrict__ CC,
    const float* __restrict__ S1, const float* __restrict__ S2,
    float* __restrict__ out)
{
    __shared__ float4 lds[CH * 8];
    const int j       = blockIdx.x * 256 + threadIdx.x;
    const int i0slice = blockIdx.y * ROWS_PER_SLICE;

    float sj[8], R1[8], R2[8];
    #pragma unroll
    for (int h = 0; h < 8; ++h) {
        sj[h] = CC[j * 32 + h];
        const float C1 = CC[j * 32 + 8 + h];
        const float C2 = CC[j * 32 + 16 + h];
        const float d  = S1[h * N_NODES + j] * C1 + S2[h * N_NODES + j] * C2;
        const float iv = (d > 0.0f) ? (1.0f / d) : 0.0f;
        R1[h] = C1 * iv;
        R2[h] = C2 * iv;
    }

    for (int c0 = 0; c0 < ROWS_PER_SLICE; c0 += CH) {
        const int i0 = i0slice + c0;
        __syncthreads();
        const float4* src = (const float4*)(TP + i0 * 32);
        for (int t = threadIdx.x; t < CH * 8; t += 256) lds[t] = src[t];
        __syncthreads();
        for (int ii = 0; ii < CH; ++ii) {
            const int i = i0 + ii;
            const int a = adj[i * N_NODES + j];
            float sum = 0.0f;
            #pragma unroll
            for (int h = 0; h < 8; ++h) {
                const float4 q = lds[ii * 8 + h];
                const float  v = q.x + sj[h];
                const bool pos = (v >= 0.0f);
                const float p  = pos ? q.y : q.z;
                const float rr = pos ? R1[h] : R2[h];
                sum = fmaf(p, rr, sum);
            }
            out[i * N_NODES + j] = (a > 0) ? (sum * 0.125f) : 0.0f;
        }
    }
}

// ---------------------------------------------------------------------------
extern "C" void kernel_launch(void* const* d_in, const int* in_sizes, int n_in,
                              void* d_out, int out_size, void* d_ws, size_t ws_size,
                              hipStream_t stream)
{
    (void)in_sizes; (void)n_in; (void)out_size; (void)ws_size;
    const float* x      = (const float*)d_in[0];
    const int*   adj    = (const int*)  d_in[1];
    const float* linear = (const float*)d_in[2];
    const float* asrc   = (const float*)d_in[3];
    const float* atar   = (const float*)d_in[4];
    float* out = (float*)d_out;

    // Workspace layout (floats). Total ~2.17M * ISLICES partials = ~8.7 MB.
    float* ws    = (float*)d_ws;
    const int HN = HEADS * N_NODES;
    float* s_src = ws;                         // HN
    float* s_tar = s_src + HN;                 // HN
    float* TP    = s_tar + HN;                 // 32*N
    float* CC    = TP + 32 * N_NODES;          // 32*N
    float* S1p   = CC + 32 * N_NODES;          // ISLICES*HN
    float* S2p   = S1p + ISLICES * HN;         // ISLICES*HN
    float* S1    = S2p + ISLICES * HN;         // HN
    float* S2    = S1 + HN;                    // HN
    float* M     = S2 + HN;                    // HEADS

    proj_wmma_kernel<<<N_NODES / 16, 256, 0, stream>>>(x, linear, asrc, atar,
                                                       s_src, s_tar);
    maxred_kernel<<<HEADS, 256, 0, stream>>>(s_tar, M);
    pexp_kernel<<<N_NODES / 256, 256, 0, stream>>>(s_src, s_tar, M, TP, CC);

    dim3 gtile(N_NODES / 256, ISLICES);
    colsum_kernel<<<gtile, 256, 0, stream>>>(adj, TP, CC, S1p, S2p);

    dim3 gred(N_NODES / 256, HEADS);
    slice_reduce_kernel<<<gred, 256, 0, stream>>>(S1p, S2p, S1, S2);

    out_kernel<<<gtile, 256, 0, stream>>>(adj, TP, CC, S1, S2, out);
}